// SuppressionFine_6597069767428
// MI455X (gfx1250) — compile-verified
//
#include <hip/hip_runtime.h>
#include <cstdint>
#include <cmath>

// ---------------------------------------------------------------------------
// SSD-style NMS post-processing for gfx1250 (MI455X).
//   Input : y_pred (8, 8732, 102) f32
//   Stage1: 160 workgroups (image x class). Stage conf+boxes into LDS using
//           CDNA5 async global->LDS loads, then run the 400-step greedy NMS
//           with wave32 shfl_xor argmax reductions. Writes (8,8000,7) rows
//           into workspace.
//   Stage2: 8 workgroups, stable top-200 selection per image, gathers rows
//           to d_out (8,200,7).
// No matrix ops exist in this workload (IoU/argmax are min/max compares), so
// WMMA is inapplicable; the CDNA5-specific paths used are wave32 +
// global_load_async_to_lds_b32 / s_wait_asynccnt + 320KB-class LDS tiles.
// ---------------------------------------------------------------------------

#define NB       8732
#define LAST_    102
#define NCLS     21
#define NCLASSM1 20
#define TOPK     200
#define NMSOUT   400
#define NEGV     (-1000000000.0f)
#define IOU_T    0.45f
#define CONF_T   0.5f
#define SPAD     8736              // padded per-array stride in LDS floats

// CDNA5 async data mover: copies the float at global address gptr to LDS offset lds_off, tracked by ASYNCcnt.
__device__ __forceinline__ void async_g2l(uint32_t lds_off, const float* gptr) {
    asm volatile("global_load_async_to_lds_b32 %0, %1, off"
                 :: "v"(lds_off), "v"(gptr) : "memory");
}
__device__ __forceinline__ void wait_async_all() {
    asm volatile("s_wait_asynccnt 0x0" ::: "memory");
}
__device__ __forceinline__ uint32_t lds_off_of(const void* p) {
    // generic LDS address = aperture | 32-bit LDS offset; low 32 bits are the offset
    return (uint32_t)(uintptr_t)p;
}

// wave32 argmax reduce: max value, first (smallest) index on ties
__device__ __forceinline__ void wave_argmax(float& v, int& ix) {
#pragma unroll
    for (int off = 16; off > 0; off >>= 1) {
        float ov = __shfl_xor(v, off, 32);
        int   oi = __shfl_xor(ix, off, 32);
        if (ov > v || (ov == v && oi < ix)) { v = ov; ix = oi; }
    }
}

__global__ __launch_bounds__(256)
void nms_per_class_kernel(const float* __restrict__ yp, float* __restrict__ filt) {
    extern __shared__ float sm[];
    float* sc  = sm;                 // [SPAD] live scores
    float* lx1 = sm + SPAD;          // [SPAD] box coords (SoA, conflict-free)
    float* ly1 = sm + 2 * SPAD;
    float* lx2 = sm + 3 * SPAD;
    float* ly2 = sm + 4 * SPAD;
    float* rv  = sm + 5 * SPAD;      // [16] per-wave partial max values
    float* bb  = rv + 16;            // [8]  broadcast: bval, bx1,by1,bx2,by2
    int*   ri  = (int*)(bb + 8);     // [16] per-wave partial max indices
    int*   bip = ri + 16;            // [1]  broadcast best index

    const int tid = threadIdx.x;
    const int blk = blockIdx.x;
    const int b   = blk / NCLASSM1;
    const int c   = blk % NCLASSM1;
    const float* base = yp + (size_t)b * NB * LAST_;

    // ---- stage: conf (threshold applied) via VALU path, boxes via async TDM path
    for (int i = tid; i < NB; i += 256) {
        const float* row = base + (size_t)i * LAST_;
        float conf = row[1 + c];
        sc[i] = conf > CONF_T ? conf : NEGV;
        const float* bx = row + NCLS + 4 * c;
        async_g2l(lds_off_of(&lx1[i]), bx + 0);
        async_g2l(lds_off_of(&ly1[i]), bx + 1);
        async_g2l(lds_off_of(&lx2[i]), bx + 2);
        async_g2l(lds_off_of(&ly2[i]), bx + 3);
    }
    wait_async_all();
    __syncthreads();

    float* outbase = filt + ((size_t)b * 8000 + (size_t)c * NMSOUT) * 7;

    for (int t = 0; t < NMSOUT; ++t) {
        // ---- argmax over live scores (first-index tie-break)
        float v = -INFINITY; int ix = 0x7fffffff;
        for (int i = tid; i < NB; i += 256) {
            float s = sc[i];
            if (s > v) { v = s; ix = i; }      // per-thread indices ascend -> keeps first
        }
        wave_argmax(v, ix);
        const int wave = tid >> 5;
        if ((tid & 31) == 0) { rv[wave] = v; ri[wave] = ix; }
        __syncthreads();
        if (tid == 0) {
            float bv = rv[0]; int bx = ri[0];
#pragma unroll
            for (int w = 1; w < 8; ++w) {
                float wv = rv[w]; int wi = ri[w];
                if (wv > bv || (wv == bv && wi < bx)) { bv = wv; bx = wi; }
            }
            bb[0] = bv; bip[0] = bx;
            bb[1] = lx1[bx]; bb[2] = ly1[bx]; bb[3] = lx2[bx]; bb[4] = ly2[bx];
        }
        __syncthreads();

        const float bv   = bb[0];
        const int   bidx = bip[0];
        const bool  valid = bv > (0.5f * NEGV);
        if (!valid) {
            // all remaining selections invalid -> zero rows t..399, done
            const int rem = (NMSOUT - t) * 7;
            float* zp = outbase + (size_t)t * 7;
            for (int e = tid; e < rem; e += 256) zp[e] = 0.0f;
            break;
        }
        const float bx1 = bb[1], by1 = bb[2], bx2 = bb[3], by2 = bb[4];

        if (tid == 0) {
            float* orow = outbase + (size_t)t * 7;
            orow[0] = (float)(c + 1);
            orow[1] = bv;
            orow[2] = bx1; orow[3] = by1; orow[4] = bx2; orow[5] = by2;
            orow[6] = base[(size_t)bidx * LAST_ + (LAST_ - 1)];   // extra column
        }

        // ---- IoU suppression (division-free: iou>T <=> inter > T*union, union>0)
        const float areaA = (bx2 - bx1) * (by2 - by1);
        for (int i = tid; i < NB; i += 256) {
            float ax1 = lx1[i], ay1 = ly1[i], ax2 = lx2[i], ay2 = ly2[i];
            float iw = fminf(bx2, ax2) - fmaxf(bx1, ax1);
            float ih = fminf(by2, ay2) - fmaxf(by1, ay1);
            float inter = fmaxf(iw, 0.0f) * fmaxf(ih, 0.0f);
            float areaB = (ax2 - ax1) * (ay2 - ay1);
            float uni   = areaA + areaB - inter;
            bool  sup   = (i == bidx) || ((uni > 0.0f) && (inter > IOU_T * uni));
            if (sup) sc[i] = NEGV;
        }
        __syncthreads();
    }
}

__global__ __launch_bounds__(256)
void topk_gather_kernel(const float* __restrict__ filt, float* __restrict__ out) {
    __shared__ float sc[8000];
    __shared__ float rv[8];
    __shared__ int   ri[8];
    __shared__ int   bsel;

    const int tid = threadIdx.x;
    const int b   = blockIdx.x;
    const float* fb = filt + (size_t)b * 8000 * 7;

    // stage the 8000 score column (stride-7 gather) via async LDS loads
    for (int j = tid; j < 8000; j += 256)
        async_g2l(lds_off_of(&sc[j]), fb + (size_t)j * 7 + 1);
    wait_async_all();
    __syncthreads();

    float* ob = out + (size_t)b * TOPK * 7;
    for (int k = 0; k < TOPK; ++k) {
        float v = -INFINITY; int ix = 0x7fffffff;
        for (int j = tid; j < 8000; j += 256) {
            float s = sc[j];
            if (s > v) { v = s; ix = j; }
        }
        wave_argmax(v, ix);
        if ((tid & 31) == 0) { rv[tid >> 5] = v; ri[tid >> 5] = ix; }
        __syncthreads();
        if (tid == 0) {
            float bv = rv[0]; int bx = ri[0];
#pragma unroll
            for (int w = 1; w < 8; ++w) {
                if (rv[w] > bv || (rv[w] == bv && ri[w] < bx)) { bv = rv[w]; bx = ri[w]; }
            }
            bsel = bx;
            sc[bx] = -INFINITY;       // remove from future rounds (stable top-k)
        }
        __syncthreads();
        const int sel = bsel;
        if (tid < 7)
            ob[(size_t)k * 7 + tid] = fb[(size_t)sel * 7 + tid];
        // no extra barrier needed: bsel/rv/ri rewrites are fenced by the two
        // barriers of the next iteration.
    }
}

extern "C" void kernel_launch(void* const* d_in, const int* in_sizes, int n_in,
                              void* d_out, int out_size, void* d_ws, size_t ws_size,
                              hipStream_t stream) {
    (void)in_sizes; (void)n_in; (void)out_size; (void)ws_size;
    const float* yp   = (const float*)d_in[0];
    float*       filt = (float*)d_ws;           // (8, 8000, 7) f32 = 1.79 MB
    float*       out  = (float*)d_out;          // (8, 200, 7) f32

    const size_t dynsmem = (size_t)(5 * SPAD + 16 + 8 + 16 + 8) * sizeof(float);
    nms_per_class_kernel<<<8 * NCLASSM1, 256, dynsmem, stream>>>(yp, filt);
    topk_gather_kernel<<<8, 256, 0, stream>>>(filt, out);
}